// SwitchableLayerNorm_41901700939886
// MI455X (gfx1250) — compile-verified
//
#include <hip/hip_runtime.h>

typedef __attribute__((ext_vector_type(2))) float v2f;
typedef __attribute__((ext_vector_type(4))) float v4f;
typedef __attribute__((ext_vector_type(8))) float v8f;

#define DIM 768
#define NK 16
#define WAVES_PER_BLOCK 8
#define ROWS_PER_WAVE 16
#define ROWS_PER_BLOCK (WAVES_PER_BLOCK * ROWS_PER_WAVE)  // 128

__global__ __launch_bounds__(256) void sln_fused_kernel(
    const float* __restrict__ x,
    const float* __restrict__ weights,
    const float* __restrict__ biases,
    const float* __restrict__ centroids,
    float* __restrict__ out,      // [nrows * DIM]
    float* __restrict__ outBkt)   // [nrows] bucket ids as float
{
  __shared__ float sMean[WAVES_PER_BLOCK][ROWS_PER_WAVE];
  __shared__ float sRstd[WAVES_PER_BLOCK][ROWS_PER_WAVE];
  __shared__ int   sBkt [WAVES_PER_BLOCK][ROWS_PER_WAVE];

  const int tid     = threadIdx.x;
  const int w       = tid >> 5;      // wave in block
  const int lane    = tid & 31;
  const int half    = lane >> 4;     // 0: lanes 0-15, 1: lanes 16-31
  const int l16     = lane & 15;
  const int rowbase = (blockIdx.x * WAVES_PER_BLOCK + w) * ROWS_PER_WAVE;

  // ---------------- Phase 1: WMMA xc accumulation + LN stats (one x pass) ---
  // A (16x4 f32): lane l16 = row M, half selects K={0,1} vs K={2,3}
  // B (4x16 f32): lane l16 = centroid N, same K mapping -> b64 loads each
  const float* xr = x         + (size_t)(rowbase + l16) * DIM + half * 2;
  const float* cr = centroids + (size_t)l16            * DIM + half * 2;

  v8f acc = {0.f, 0.f, 0.f, 0.f, 0.f, 0.f, 0.f, 0.f};
  float s = 0.f, sq = 0.f, cs = 0.f;

#pragma unroll 4
  for (int i = 0; i < DIM / 4; ++i) {
    v2f a = *(const v2f*)(xr + 4 * i);
    v2f b = *(const v2f*)(cr + 4 * i);
    s  += a.x + a.y;
    sq += a.x * a.x + a.y * a.y;
    cs += b.x * b.x + b.y * b.y;
    // D = A(16x4) * B(4x16) + C : v_wmma_f32_16x16x4_f32
    acc = __builtin_amdgcn_wmma_f32_16x16x4_f32(
        /*neg_a=*/false, a, /*neg_b=*/false, b,
        /*c_mod=*/(short)0, acc, /*reuse_a=*/false, /*reuse_b=*/false);
  }

  // combine half-wave partials: lanes l and l^16 cover complementary K subsets
  s  += __shfl_xor(s,  16, 32);
  sq += __shfl_xor(sq, 16, 32);
  cs += __shfl_xor(cs, 16, 32);    // cs = |centroid[l16]|^2 (complete)

  const float inv_d = 1.0f / (float)DIM;
  const float mean  = s * inv_d;
  const float var   = sq * inv_d - mean * mean;
  const float rstd  = rsqrtf(var + 1e-5f);
  if (lane < 16) { sMean[w][lane] = mean; sRstd[w][lane] = rstd; }

  // ---------------- Phase 2: argmin_k (|c_k|^2 - 2 x.c_k) per row -----------
  // C layout: VGPR r, lanes 0-15 -> M=r, N=lane; lanes 16-31 -> M=r+8, N=lane-16
#pragma unroll
  for (int r = 0; r < 8; ++r) {
    float val = cs - 2.0f * acc[r];
    int   id  = l16;
#pragma unroll
    for (int off = 1; off < 16; off <<= 1) {
      float ov = __shfl_xor(val, off, 32);
      int   oi = __shfl_xor(id,  off, 32);
      if (ov < val || (ov == val && oi < id)) { val = ov; id = oi; }
    }
    if (l16 == 0) sBkt[w][r + half * 8] = id;
  }
  __syncthreads();

  if (lane < 16) outBkt[rowbase + lane] = (float)sBkt[w][lane];

  // ---------------- Phase 3: streaming normalize + per-bucket affine --------
  // x re-read: L2 hit from phase 1, last use -> NT load.
  // out: never re-read -> NT store (keeps the 96MB write stream out of L2 so
  // x stays resident between the two passes; x+out would exactly fill 192MB).
  for (int m = 0; m < ROWS_PER_WAVE; ++m) {
    const int   row = rowbase + m;
    const int   bk  = sBkt[w][m];
    const float mm  = sMean[w][m];
    const float rs  = sRstd[w][m];
    const v4f* xr4 = (const v4f*)(x       + (size_t)row * DIM);
    const v4f* wr4 = (const v4f*)(weights + (size_t)bk  * DIM);
    const v4f* br4 = (const v4f*)(biases  + (size_t)bk  * DIM);
    v4f*       or4 = (v4f*)(out + (size_t)row * DIM);
#pragma unroll
    for (int j = 0; j < DIM / (32 * 4); ++j) {   // 6 iterations of b128 traffic
      const int c = j * 32 + lane;
      const v4f xv = __builtin_nontemporal_load(&xr4[c]);  // last use of x
      const v4f wv = wr4[c];                               // hot, keep cached
      const v4f bv = br4[c];
      v4f ov;
      ov.x = (xv.x - mm) * rs * wv.x + bv.x;
      ov.y = (xv.y - mm) * rs * wv.y + bv.y;
      ov.z = (xv.z - mm) * rs * wv.z + bv.z;
      ov.w = (xv.w - mm) * rs * wv.w + bv.w;
      __builtin_nontemporal_store(ov, &or4[c]);            // streaming store
    }
  }
}

extern "C" void kernel_launch(void* const* d_in, const int* in_sizes, int n_in,
                              void* d_out, int out_size, void* d_ws, size_t ws_size,
                              hipStream_t stream) {
  const float* x         = (const float*)d_in[0];
  const float* weights   = (const float*)d_in[1];
  const float* biases    = (const float*)d_in[2];
  const float* centroids = (const float*)d_in[3];
  float* out = (float*)d_out;

  const int nrows = in_sizes[0] / DIM;              // B*S = 32768
  float* outBkt = out + (size_t)nrows * DIM;        // tuple output #2

  const int grid = nrows / ROWS_PER_BLOCK;          // 256 blocks
  sln_fused_kernel<<<grid, 256, 0, stream>>>(x, weights, biases, centroids,
                                             out, outBkt);
}